// TestModel2_30270929502340
// MI455X (gfx1250) — compile-verified
//
#include <hip/hip_runtime.h>

// TestModel2: 370-element thresholded quadratic window around the (signed)
// argmin-nearest target frequency, normalized by the global max.
//
// Latency-bound (1.5 KB total traffic -> ~60 ps of HBM time at 23.3 TB/s):
// the whole problem fits in ONE wave32. Each lane owns 12 strided elements
// (12*32 = 384 >= 370), the global max is a 5-step ds_swizzle XOR butterfly
// entirely in registers -- no LDS arrays, no barriers, no second phase.

#define FMIN_F   30.0f
#define N_OUT    370        // arange(30, 400)
#define WIDTH_F  10.0f
#define PER_LANE 12         // ceil(370/32)

// ds_swizzle group-of-32 XOR mode: offset = (xor_mask << 10) | (or=0 << 5) | and=0x1f
#define SWZ_XOR16 0x401f
#define SWZ_XOR8  0x201f
#define SWZ_XOR4  0x101f
#define SWZ_XOR2  0x081f
#define SWZ_XOR1  0x041f

#define SWZ_MAX(v, pat)                                                     \
    v = fmaxf(v, __int_as_float(                                            \
            __builtin_amdgcn_ds_swizzle(__float_as_int(v), (pat))))

__global__ __launch_bounds__(32, 1)
void freq_window_norm_kernel(const float* __restrict__ xin,
                             const float* __restrict__ yin,
                             float* __restrict__ out) {
    const float fx = xin[0];   // uniform scalar loads (s_load path)
    const float fy = yin[0];
    const int lane = threadIdx.x;   // wave32: lane == tid

    float val[PER_LANE];
    float m = 0.0f;

    #pragma unroll
    for (int j = 0; j < PER_LANE; ++j) {
        const int   i  = j * 32 + lane;        // strided -> coalesced stores
        const float xs = FMIN_F + (float)i;
        const float d0 = fx - xs;              // signed diffs (no abs!)
        const float d1 = fy - xs;
        // jnp.argmin: first index wins on tie -> pick fy only if d1 < d0
        const float f  = (d1 < d0) ? fy : fx;
        const float e  = xs - f;
        const float d  = e * e;
        const float w  = WIDTH_F - d;
        float yv = (d > WIDTH_F) ? 0.0f : w * w;  // strict '>' as reference
        if (i >= N_OUT) yv = 0.0f;  // xs >= 400 is outside reference range
        val[j] = yv;
        m = fmaxf(m, yv);
    }

    // Wave-wide max: 5 single-instruction XOR butterflies, all lanes active.
    SWZ_MAX(m, SWZ_XOR16);
    SWZ_MAX(m, SWZ_XOR8);
    SWZ_MAX(m, SWZ_XOR4);
    SWZ_MAX(m, SWZ_XOR2);
    SWZ_MAX(m, SWZ_XOR1);

    // One full-precision divide instead of 370; 0 * inf = NaN matches jnp 0/0.
    const float inv = 1.0f / m;

    #pragma unroll
    for (int j = 0; j < PER_LANE; ++j) {
        const int i = j * 32 + lane;
        if (i < N_OUT) out[i] = val[j] * inv;
    }
}

extern "C" void kernel_launch(void* const* d_in, const int* in_sizes, int n_in,
                              void* d_out, int out_size, void* d_ws, size_t ws_size,
                              hipStream_t stream) {
    (void)in_sizes; (void)n_in; (void)d_ws; (void)ws_size; (void)out_size;
    const float* x = (const float*)d_in[0];
    const float* y = (const float*)d_in[1];
    float* out     = (float*)d_out;
    // Entire problem in one wave32: zero barriers, zero LDS traffic.
    freq_window_norm_kernel<<<1, 32, 0, stream>>>(x, y, out);
}